// MaskHead_56203942035882
// MI455X (gfx1250) — compile-verified
//
#include <hip/hip_runtime.h>

// CDNA5 (gfx1250) WMMA fragment types
typedef __attribute__((ext_vector_type(2))) float v2f;
typedef __attribute__((ext_vector_type(8))) float v8f;
typedef __attribute__((ext_vector_type(4))) int   v4i;

// async-copy pointer types: builtin wants (int4 AS1*, int4 AS3*, imm, imm)
typedef __attribute__((address_space(1))) v4i* g_v4i_ptr;
typedef __attribute__((address_space(3))) v4i* l_v4i_ptr;

#define BATCH 8
#define NINST 100
#define CDIM  256
#define HW    40000
#define KC    64           // K chunk staged in LDS
#define NT    32           // HW columns per workgroup (two 16-wide WMMA tiles)
#define APAD  136          // A LDS row stride: 2*136 mod 64 = 16 -> half-waves hit disjoint banks
#define BPAD  40           // B LDS row stride: 2*40  mod 64 = 16 -> half-waves hit disjoint banks
#define NCHUNK (CDIM / KC) // 4

// ---- gfx1250 async global->LDS copy (ASYNCcnt-tracked), with sync fallback ----
#if __has_builtin(__builtin_amdgcn_global_load_async_to_lds_b128)
#define ASYNC_B128(gsrc, ldst) \
    __builtin_amdgcn_global_load_async_to_lds_b128( \
        (g_v4i_ptr)(unsigned long long)(gsrc), \
        (l_v4i_ptr)(unsigned int)(unsigned long long)(ldst), 0, 0)
#else
#define ASYNC_B128(gsrc, ldst) (*(float4*)(ldst) = *(const float4*)(gsrc))
#endif

#if __has_builtin(__builtin_amdgcn_s_wait_asynccnt)
#define WAIT_ASYNC(n) __builtin_amdgcn_s_wait_asynccnt(n)
#elif defined(__gfx1250__)
#define WAIT_ASYNC(n) asm volatile("s_wait_asynccnt " #n ::: "memory")
#else
#define WAIT_ASYNC(n)
#endif

__global__ __launch_bounds__(256)
void maskhead_wmma_f32(const float* __restrict__ feats,   // [B, C, H*W]
                       const float* __restrict__ kerns,   // [B, N, C]
                       float* __restrict__ out) {         // [B, N, H*W]
    __shared__ float Alds[KC][APAD];        // A chunk transposed: Alds[k][m], m padded to 128
    __shared__ float Blds[2][KC][BPAD];     // double-buffered B chunk: Blds[buf][k][n], n<NT

    const int wg   = blockIdx.x;                 // 0 .. BATCH*(HW/NT)-1
    const int b    = wg / (HW / NT);
    const int col0 = (wg % (HW / NT)) * NT;

    const int tid  = threadIdx.x;
    const int lane = tid & 31;
    const int wave = tid >> 5;                   // 0..7 -> M tile index

    const float* fB = feats + (size_t)b * CDIM * HW;
    const float* kB = kerns + (size_t)b * NINST * CDIM;

    // B-staging mapping: 512 x b128 per chunk, 2 per thread
    const int bk0 = tid >> 3;                    // 0..31  (row of first b128)
    const int bs  = tid & 7;                     // 0..7   (16B segment in row)

    v8f acc0 = {};
    v8f acc1 = {};

    // ---- issue async B loads for chunk 0 into buffer 0 ----
    #pragma unroll
    for (int i = 0; i < 2; ++i) {
        int k = bk0 + i * 32;
        ASYNC_B128(fB + (size_t)k * HW + col0 + bs * 4, &Blds[0][k][bs * 4]);
    }

    for (int kci = 0; kci < NCHUNK; ++kci) {
        const int kc = kci * KC;

        // ---- stage A chunk: rows 0..127 (>=100 zero), transposed into Alds[k][m] ----
        // wave-contiguous m => conflict-free ds stores; 8 float4 per thread
        #pragma unroll
        for (int i = 0; i < 8; ++i) {
            int idx = tid + i * 256;             // 0..2047
            int m   = idx & 127;
            int s   = idx >> 7;                  // 0..15 float4 segment within KC
            float4 v = make_float4(0.f, 0.f, 0.f, 0.f);
            if (m < NINST) {
                v = *(const float4*)(kB + m * CDIM + kc + s * 4);
            }
            int k = s * 4;
            Alds[k + 0][m] = v.x;
            Alds[k + 1][m] = v.y;
            Alds[k + 2][m] = v.z;
            Alds[k + 3][m] = v.w;
        }

        // ---- issue async B loads for next chunk into the other buffer ----
        if (kci + 1 < NCHUNK) {
            const float* fnext = fB + (size_t)(kc + KC) * HW;
            #pragma unroll
            for (int i = 0; i < 2; ++i) {
                int k = bk0 + i * 32;
                ASYNC_B128(fnext + (size_t)k * HW + col0 + bs * 4,
                           &Blds[(kci + 1) & 1][k][bs * 4]);
            }
            // prefetch next A chunk (L2-resident, keeps it warm)
            __builtin_prefetch(kB + (tid & 127) * CDIM + kc + KC, 0, 1);
            WAIT_ASYNC(2);                       // current chunk's B done; next still in flight
        } else {
            WAIT_ASYNC(0);                       // drain all async B
        }
        __syncthreads();                         // A stores + everyone's B visible

        // ---- compute: 32 x v_wmma_f32_16x16x4_f32 per chunk ----
        // ISA layouts (wave32):
        //   A 16x4 : lane%16 = M, lane/16 selects {K,K+1} vs {K+2,K+3}
        //   B 4x16 : lane%16 = N, same K-half split
        //   D 16x16: VGPR r -> M = r + 8*(lane/16), N = lane%16
        const float (*Bbuf)[BPAD] = Blds[kci & 1];
        const int mfrag = wave * 16 + (lane & 15);
        const int khalf = (lane >> 4) << 1;      // 0 or 2
        const int nfrag = lane & 15;
        #pragma unroll
        for (int k = 0; k < KC; k += 4) {
            v2f a, b0, b1;
            a.x  = Alds[k + khalf + 0][mfrag];
            a.y  = Alds[k + khalf + 1][mfrag];
            b0.x = Bbuf[k + khalf + 0][nfrag];
            b0.y = Bbuf[k + khalf + 1][nfrag];
            b1.x = Bbuf[k + khalf + 0][16 + nfrag];
            b1.y = Bbuf[k + khalf + 1][16 + nfrag];
            acc0 = __builtin_amdgcn_wmma_f32_16x16x4_f32(
                false, a, false, b0, (short)0, acc0, false, false);
            acc1 = __builtin_amdgcn_wmma_f32_16x16x4_f32(
                false, a, false, b1, (short)0, acc1, false, false);
        }
        __syncthreads();                         // done reading buf before it is overwritten
    }

    // ---- store D tiles: rows >= NINST were zero-padded compute, drop them ----
    const int mrow0 = wave * 16 + (lane >> 4) * 8;
    const int nfrag = lane & 15;
    float* oB = out + (size_t)b * NINST * HW + col0 + nfrag;
    #pragma unroll
    for (int r = 0; r < 8; ++r) {
        int m = mrow0 + r;
        if (m < NINST) {
            oB[(size_t)m * HW]      = acc0[r];
            oB[(size_t)m * HW + 16] = acc1[r];
        }
    }
}

extern "C" void kernel_launch(void* const* d_in, const int* in_sizes, int n_in,
                              void* d_out, int out_size, void* d_ws, size_t ws_size,
                              hipStream_t stream) {
    const float* feats = (const float*)d_in[0];   // [8, 256, 200, 200]
    const float* kerns = (const float*)d_in[1];   // [8, 100, 256]
    float* out = (float*)d_out;                   // [8, 100, 200, 200]

    dim3 grid(BATCH * (HW / NT));                 // 8 * 1250 = 10000 workgroups
    dim3 block(256);                              // 8 wave32
    maskhead_wmma_f32<<<grid, block, 0, stream>>>(feats, kerns, out);
}